// HDRDecoder_14955076124841
// MI455X (gfx1250) — compile-verified
//
#include <hip/hip_runtime.h>
#include <hip/hip_bf16.h>

typedef __attribute__((ext_vector_type(2))) float v2f;
typedef __attribute__((ext_vector_type(8))) float v8f;

#define WMMA_F32X4(a, b, c) \
  __builtin_amdgcn_wmma_f32_16x16x4_f32(false, (a), false, (b), (short)0, (c), false, false)

// ---------------------------------------------------------------------------
// concat z_enc (N,4,64,64) + z (N,4,64,64) -> (N,8,64,64)
// ---------------------------------------------------------------------------
__global__ void k_concat(const float* __restrict__ z_enc, const float* __restrict__ z,
                         float* __restrict__ out, int n_total) {
  int i = blockIdx.x * blockDim.x + threadIdx.x;
  if (i >= n_total) return;
  int p = i & 4095;        // 64*64
  int t = i >> 12;
  int c = t & 7;
  int n = t >> 3;
  out[i] = (c < 4) ? z_enc[(n * 4 + c) * 4096 + p]
                   : z[(n * 4 + (c - 4)) * 4096 + p];
}

// ---------------------------------------------------------------------------
// Generic 3x3 conv (pad=1) as implicit GEMM on V_WMMA_F32_16X16X4_F32.
// One workgroup per (n, 16-pixel tile). Threads cooperatively build the
// im2col panel sB[16 px][K] in LDS (row stride K+4 -> ds_load_b64 is
// bank-conflict-free across the wave). One wave per 16-output-channel tile:
//   D[oc][px] = sum_k wgt[oc][k] * im2col[k][px]
// Weight rows are contiguous float2 (K order ic*9+kh*3+kw == OIHW flat).
// ---------------------------------------------------------------------------
__global__ void k_conv_wmma(const float* __restrict__ in, const float* __restrict__ wgt,
                            const float* __restrict__ bias, float* __restrict__ out,
                            int Cin, int Hin, int Win,
                            int Cout, int Hout, int Wout,
                            int stride, int doRelu, int nPixTiles) {
  __shared__ __align__(16) float sB[16 * (576 + 4)];   // K<=576, padded rows

  const int K = Cin * 9;
  const int KP = K + 4;
  const int tile = blockIdx.x % nPixTiles;
  const int n = blockIdx.x / nPixTiles;
  const int tid = threadIdx.x;
  const int nthr = blockDim.x;

  // --- build im2col panel for this 16-pixel tile ---
  const float* ib = in + n * Cin * Hin * Win;
  for (int p = 0; p < 16; ++p) {
    int px = tile * 16 + p;
    int oy = px / Wout;
    int ox = px % Wout;
    for (int k = tid; k < K; k += nthr) {
      int ic = k / 9;
      int r9 = k - ic * 9;
      int kh = r9 / 3;
      int kw = r9 - kh * 3;
      int iy = oy * stride + kh - 1;
      int ix = ox * stride + kw - 1;
      float v = 0.0f;
      if (iy >= 0 && iy < Hin && ix >= 0 && ix < Win)
        v = ib[(ic * Hin + iy) * Win + ix];
      sB[p * KP + k] = v;
    }
  }
  __syncthreads();

  // --- one wave per oc tile ---
  const int wave = tid >> 5;
  const int lane = tid & 31;
  const int half = lane >> 4;
  const int lr = lane & 15;
  const int ocbase = wave * 16;

  int wr = ocbase + lr;
  if (wr > Cout - 1) wr = Cout - 1;            // clamp (only Cout=8 case)
  const float* wrow = wgt + wr * K;
  const float* brow = sB + lr * KP;

  v8f c = {};
  for (int k0 = 0; k0 < K; k0 += 4) {
    int kk = k0 + 2 * half;
    float2 av = *(const float2*)(wrow + kk);
    float2 bv = *(const float2*)(brow + kk);
    v2f a; a[0] = av.x; a[1] = av.y;
    v2f b; b[0] = bv.x; b[1] = bv.y;
    c = WMMA_F32X4(a, b, c);
  }

  const int px = tile * 16 + lr;               // D column = pixel
  float* ob = out + (n * Cout) * (Hout * Wout) + px;
  #pragma unroll
  for (int j = 0; j < 8; ++j) {
    int oc = ocbase + j + 8 * half;            // D row = output channel
    if (oc < Cout) {
      float v = c[j] + (bias ? bias[oc] : 0.0f);
      if (doRelu) v = fmaxf(v, 0.0f);
      ob[oc * (Hout * Wout)] = v;
    }
  }
}

// ---------------------------------------------------------------------------
// FC layer via V_WMMA_F32_16X16X4_F32 (batch rows = M dim, one wave/16 feats)
// ---------------------------------------------------------------------------
__global__ void k_fc_wmma(const float* __restrict__ in,   // [nrows x K]
                          const float* __restrict__ w,    // [Mfeat x K]
                          const float* __restrict__ bias, // [Mfeat] or null
                          float* __restrict__ out,        // [nrows x Mfeat]
                          int K, int Mfeat, int nrows, int doRelu) {
  int tile = blockIdx.x;          // blockDim.x == 32
  int lane = threadIdx.x;
  int half = lane >> 4;
  int lr = lane & 15;
  float am = (lr < nrows) ? 1.0f : 0.0f;
  int arow = (lr < nrows) ? lr : 0;

  const float* irow = in + arow * K;
  const float* wrow = w + (tile * 16 + lr) * K;

  v8f c = {};
  for (int k0 = 0; k0 < K; k0 += 4) {
    int kk = k0 + 2 * half;
    float2 av = *(const float2*)(irow + kk);
    float2 bv = *(const float2*)(wrow + kk);
    v2f a; a[0] = av.x * am; a[1] = av.y * am;
    v2f b; b[0] = bv.x;      b[1] = bv.y;
    c = WMMA_F32X4(a, b, c);
  }

  int of = tile * 16 + lr;
  float bval = bias ? bias[of] : 0.0f;
  #pragma unroll
  for (int j = 0; j < 8; ++j) {
    int m = j + 8 * half;
    if (m < nrows) {
      float v = c[j] + bval;
      if (doRelu) v = fmaxf(v, 0.0f);
      out[m * Mfeat + of] = v;
    }
  }
}

// ---------------------------------------------------------------------------
// pred 1x1 conv as GEMM (WMMA), fusing relu(l2 + g_broadcast):
//   grid[n][z][gy][gx][ch] = sum_k pw[oc][k]*relu(l2o[n][k][px]+g[n][k]) + pb
// ---------------------------------------------------------------------------
__global__ void k_pred_wmma(const float* __restrict__ l2o,  // [N][64][256]
                            const float* __restrict__ gvec, // [N][64]
                            const float* __restrict__ pw,   // [96][64]
                            const float* __restrict__ pb,   // [96]
                            float* __restrict__ bgrid) {    // [N][8][16][16][12]
  int wid = blockIdx.x;               // blockDim.x == 32
  int ntile = wid % 16; wid /= 16;
  int mtile = wid % 6;  int n = wid / 6;
  int lane = threadIdx.x;
  int half = lane >> 4;
  int lr = lane & 15;

  const float* wrow = pw + (mtile * 16 + lr) * 64;
  const float* bbase = l2o + n * 64 * 256 + ntile * 16 + lr;
  const float* gb = gvec + n * 64;

  v8f c = {};
  for (int k0 = 0; k0 < 64; k0 += 4) {
    int kk = k0 + 2 * half;
    float2 av = *(const float2*)(wrow + kk);
    v2f a; a[0] = av.x; a[1] = av.y;
    v2f b;
    b[0] = fmaxf(bbase[kk * 256] + gb[kk], 0.0f);
    b[1] = fmaxf(bbase[(kk + 1) * 256] + gb[kk + 1], 0.0f);
    c = WMMA_F32X4(a, b, c);
  }

  int px = ntile * 16 + lr;           // gy*16+gx
  int gy = px >> 4, gx = px & 15;
  #pragma unroll
  for (int j = 0; j < 8; ++j) {
    int oc = mtile * 16 + j + 8 * half;
    int ch = oc >> 3;                  // (N,96)->(N,12,8): oc = ch*8+z
    int z  = oc & 7;
    bgrid[(((n * 8 + z) * 16 + gy) * 16 + gx) * 12 + ch] = c[j] + pb[oc];
  }
}

// ---------------------------------------------------------------------------
// Fused guide + trilinear slice + affine apply. One workgroup per (row h, n).
// Grid is pre-lerped along y into LDS [z][gx][12] (6 KB); each pixel needs
// 4 LDS corner gathers (3x ds_load_b128). Guidance/output moved as float4:
// 3 global_load_b128 + 3 global_store_b128 per thread (4 px), so the kernel
// runs at the mandatory 24 B/px HBM traffic.
// ---------------------------------------------------------------------------
__global__ void k_slice_apply(const float* __restrict__ bgrid,    // [4][8][16][16][12]
                              const float* __restrict__ guidance, // [4][3][1024][1024]
                              const float* __restrict__ Mm,       // [3][3]
                              const float* __restrict__ Mb,       // [3]
                              const float* __restrict__ thr,      // [3][16]
                              const float* __restrict__ slp,      // [3][16]
                              const float* __restrict__ gbias,    // [1]
                              float* __restrict__ out) {          // [4][3][1024][1024]
  const int H = 1024, W = 1024;
  int h = blockIdx.x;
  int n = blockIdx.y;
  int tid = threadIdx.x;              // 256 threads

  __shared__ __align__(16) float sG[8 * 16 * 12];   // y-lerped grid row
  __shared__ float sPar[112];

  // row -> grid y coordinate
  float hh = (float)h * (2.0f / 1023.0f) - 1.0f;
  float gyc = fminf(fmaxf(((hh + 1.0f) * 16.0f - 1.0f) * 0.5f, 0.0f), 15.0f);
  float gy0f = floorf(gyc);
  float fy = gyc - gy0f;
  int iy0 = min(max((int)gy0f, 0), 15);
  int iy1 = min(iy0 + 1, 15);
  float wy0 = 1.0f - fy, wy1 = fy;

  for (int i = tid; i < 109; i += 256) {
    float v;
    if (i < 9)        v = Mm[i];
    else if (i < 12)  v = Mb[i - 9];
    else if (i < 60)  v = thr[i - 12];
    else if (i < 108) v = slp[i - 60];
    else              v = gbias[0];
    sPar[i] = v;
  }

  const float* gn = bgrid + n * (8 * 16 * 16 * 12);
  for (int f4 = tid; f4 < 384; f4 += 256) {   // 1536 floats as float4
    int idx = f4 * 4;
    int z = idx / 192;                         // 16*12 floats per (z,row)
    int r = idx % 192;
    float4 a = *(const float4*)(gn + z * 3072 + iy0 * 192 + r);
    float4 b = *(const float4*)(gn + z * 3072 + iy1 * 192 + r);
    float4 v;
    v.x = a.x * wy0 + b.x * wy1;  v.y = a.y * wy0 + b.y * wy1;
    v.z = a.z * wy0 + b.z * wy1;  v.w = a.w * wy0 + b.w * wy1;
    *(float4*)(sG + idx) = v;
  }
  __syncthreads();

  const float M00 = sPar[0], M01 = sPar[1], M02 = sPar[2];
  const float M10 = sPar[3], M11 = sPar[4], M12 = sPar[5];
  const float M20 = sPar[6], M21 = sPar[7], M22 = sPar[8];
  const float B0 = sPar[9], B1 = sPar[10], B2 = sPar[11];
  const float gB = sPar[108];

  const size_t HW = (size_t)H * W;
  const float* gr  = guidance + (size_t)n * 3 * HW + (size_t)h * W;
  const float* gg  = gr + HW;
  const float* gb2 = gg + HW;
  float* o0 = out + (size_t)n * 3 * HW + (size_t)h * W;
  float* o1 = o0 + HW;
  float* o2 = o1 + HW;

  const int wbase = tid * 4;                   // 256 thr * 4 px = 1024 = W
  float4 r4 = *(const float4*)(gr + wbase);
  float4 g4 = *(const float4*)(gg + wbase);
  float4 b4 = *(const float4*)(gb2 + wbase);
  float rv[4] = {r4.x, r4.y, r4.z, r4.w};
  float gv[4] = {g4.x, g4.y, g4.z, g4.w};
  float bv[4] = {b4.x, b4.y, b4.z, b4.w};
  float ov0[4], ov1[4], ov2[4];

  #pragma unroll
  for (int i = 0; i < 4; ++i) {
    int w0 = wbase + i;
    float r = rv[i], g = gv[i], b = bv[i];

    // guide = clip( sum_c sum_t slp[c][t]*relu((rgb@M+Mb)_c - thr[c][t]) /3 + gB )
    float x0 = fmaf(r, M00, fmaf(g, M10, fmaf(b, M20, B0)));
    float x1 = fmaf(r, M01, fmaf(g, M11, fmaf(b, M21, B1)));
    float x2 = fmaf(r, M02, fmaf(g, M12, fmaf(b, M22, B2)));
    float acc = 0.0f;
    #pragma unroll
    for (int t = 0; t < 16; ++t) {
      acc = fmaf(sPar[60 + t],      fmaxf(x0 - sPar[12 + t], 0.0f), acc);
      acc = fmaf(sPar[60 + 16 + t], fmaxf(x1 - sPar[12 + 16 + t], 0.0f), acc);
      acc = fmaf(sPar[60 + 32 + t], fmaxf(x2 - sPar[12 + 32 + t], 0.0f), acc);
    }
    float guide = fminf(fmaxf(acc * (1.0f / 3.0f) + gB, 0.0f), 1.0f);

    // z coordinate
    float gzn = guide * 2.0f - 1.0f;
    float gz = fminf(fmaxf(((gzn + 1.0f) * 8.0f - 1.0f) * 0.5f, 0.0f), 7.0f);
    float gz0f = floorf(gz);
    float fz = gz - gz0f;
    int iz0 = min(max((int)gz0f, 0), 7);
    int iz1 = min(iz0 + 1, 7);

    // x coordinate
    float ww = (float)w0 * (2.0f / 1023.0f) - 1.0f;
    float gx = fminf(fmaxf(((ww + 1.0f) * 16.0f - 1.0f) * 0.5f, 0.0f), 15.0f);
    float gx0f = floorf(gx);
    float fx = gx - gx0f;
    int ix0 = min(max((int)gx0f, 0), 15);
    int ix1 = min(ix0 + 1, 15);

    float w00 = (1.0f - fz) * (1.0f - fx);
    float w01 = (1.0f - fz) * fx;
    float w10 = fz * (1.0f - fx);
    float w11 = fz * fx;

    const float4* P00 = (const float4*)(sG + (iz0 * 16 + ix0) * 12);
    const float4* P01 = (const float4*)(sG + (iz0 * 16 + ix1) * 12);
    const float4* P10 = (const float4*)(sG + (iz1 * 16 + ix0) * 12);
    const float4* P11 = (const float4*)(sG + (iz1 * 16 + ix1) * 12);

    float s[12];
    #pragma unroll
    for (int q = 0; q < 3; ++q) {
      float4 a = P00[q], bq = P01[q], cq = P10[q], dq = P11[q];
      s[q * 4 + 0] = w00 * a.x + w01 * bq.x + w10 * cq.x + w11 * dq.x;
      s[q * 4 + 1] = w00 * a.y + w01 * bq.y + w10 * cq.y + w11 * dq.y;
      s[q * 4 + 2] = w00 * a.z + w01 * bq.z + w10 * cq.z + w11 * dq.z;
      s[q * 4 + 3] = w00 * a.w + w01 * bq.w + w10 * cq.w + w11 * dq.w;
    }

    ov0[i] = fmaf(r, s[0], fmaf(g, s[1],  fmaf(b, s[2],  s[3])));
    ov1[i] = fmaf(r, s[4], fmaf(g, s[5],  fmaf(b, s[6],  s[7])));
    ov2[i] = fmaf(r, s[8], fmaf(g, s[9],  fmaf(b, s[10], s[11])));
  }

  *(float4*)(o0 + wbase) = make_float4(ov0[0], ov0[1], ov0[2], ov0[3]);
  *(float4*)(o1 + wbase) = make_float4(ov1[0], ov1[1], ov1[2], ov1[3]);
  *(float4*)(o2 + wbase) = make_float4(ov2[0], ov2[1], ov2[2], ov2[3]);
}

// ---------------------------------------------------------------------------
// Host orchestration
// ---------------------------------------------------------------------------
extern "C" void kernel_launch(void* const* d_in, const int* in_sizes, int n_in,
                              void* d_out, int out_size, void* d_ws, size_t ws_size,
                              hipStream_t stream) {
  (void)in_sizes; (void)n_in; (void)out_size; (void)ws_size;
  // setup_inputs() flat order:
  const float* z_enc = (const float*)d_in[0];   // (4,4,64,64)
  const float* z     = (const float*)d_in[1];   // (4,4,64,64)
  const float* guid  = (const float*)d_in[2];   // (4,3,1024,1024)
  const float* s1w = (const float*)d_in[3];  const float* s1b = (const float*)d_in[4];
  const float* s2w = (const float*)d_in[5];  const float* s2b = (const float*)d_in[6];
  const float* s3w = (const float*)d_in[7];  const float* s3b = (const float*)d_in[8];
  const float* s4w = (const float*)d_in[9];  const float* s4b = (const float*)d_in[10];
  const float* g1w = (const float*)d_in[11]; const float* g1b = (const float*)d_in[12];
  const float* g2w = (const float*)d_in[13]; const float* g2b = (const float*)d_in[14];
  const float* fc3w = (const float*)d_in[15]; const float* fc3b = (const float*)d_in[16];
  const float* fc4w = (const float*)d_in[17]; const float* fc4b = (const float*)d_in[18];
  const float* fc5w = (const float*)d_in[19]; const float* fc5b = (const float*)d_in[20];
  const float* l1w = (const float*)d_in[21]; const float* l1b = (const float*)d_in[22];
  const float* l2w = (const float*)d_in[23];
  const float* pw  = (const float*)d_in[24]; const float* pb  = (const float*)d_in[25];
  const float* Mm  = (const float*)d_in[26]; const float* Mb  = (const float*)d_in[27];
  const float* thr = (const float*)d_in[28]; const float* slp = (const float*)d_in[29];
  const float* gbias = (const float*)d_in[30];
  float* out = (float*)d_out;

  // workspace layout (floats)
  float* ws = (float*)d_ws;
  float* x0    = ws;            // 131072  (4,8,64,64)
  float* x1    = x0 + 131072;   // 32768   (4,8,32,32)
  float* x2    = x1 + 32768;    // 16384   (4,16,16,16)
  float* x3    = x2 + 16384;    // 32768   (4,32,16,16)
  float* splat = x3 + 32768;    // 65536   (4,64,16,16)
  float* gb1   = splat + 65536; // 16384   (4,64,8,8)
  float* gb2   = gb1 + 16384;   // 4096    (4,64,4,4) == [4][1024]
  float* fc3o  = gb2 + 4096;    // 1024    [4][256]
  float* fc4o  = fc3o + 1024;   // 512     [4][128]
  float* fc5o  = fc4o + 512;    // 256     [4][64]
  float* l1o   = fc5o + 256;    // 65536   (4,64,16,16)
  float* l2o   = l1o + 65536;   // 65536   (4,64,16,16)
  float* bgrid = l2o + 65536;   // 98304   [4][8][16][16][12]

  // --- coefficient network (all convs on WMMA implicit GEMM) ---
  k_concat<<<512, 256, 0, stream>>>(z_enc, z, x0, 131072);
  //                blocks            threads           Cin Hin Win Cout Hout Wout st relu tiles
  k_conv_wmma<<<4 * 64, 32,  0, stream>>>(x0,    s1w, s1b, x1,    8, 64, 64,  8, 32, 32, 2, 1, 64);
  k_conv_wmma<<<4 * 16, 32,  0, stream>>>(x1,    s2w, s2b, x2,    8, 32, 32, 16, 16, 16, 2, 1, 16);
  k_conv_wmma<<<4 * 16, 64,  0, stream>>>(x2,    s3w, s3b, x3,   16, 16, 16, 32, 16, 16, 1, 1, 16);
  k_conv_wmma<<<4 * 16, 128, 0, stream>>>(x3,    s4w, s4b, splat,32, 16, 16, 64, 16, 16, 1, 1, 16);
  k_conv_wmma<<<4 * 4,  128, 0, stream>>>(splat, g1w, g1b, gb1,  64, 16, 16, 64,  8,  8, 2, 1, 4);
  k_conv_wmma<<<4 * 1,  128, 0, stream>>>(gb1,   g2w, g2b, gb2,  64,  8,  8, 64,  4,  4, 2, 1, 1);

  k_fc_wmma<<<16, 32, 0, stream>>>(gb2, fc3w, fc3b, fc3o, 1024, 256, 4, 1);
  k_fc_wmma<<<8,  32, 0, stream>>>(fc3o, fc4w, fc4b, fc4o, 256, 128, 4, 1);
  k_fc_wmma<<<4,  32, 0, stream>>>(fc4o, fc5w, fc5b, fc5o, 128, 64, 4, 0);

  k_conv_wmma<<<4 * 16, 128, 0, stream>>>(splat, l1w, l1b, l1o, 64, 16, 16, 64, 16, 16, 1, 1, 16);
  k_conv_wmma<<<4 * 16, 128, 0, stream>>>(l1o,   l2w, nullptr, l2o, 64, 16, 16, 64, 16, 16, 1, 0, 16);

  k_pred_wmma<<<4 * 6 * 16, 32, 0, stream>>>(l2o, fc5o, pw, pb, bgrid);

  // --- fused guide + slice + apply (bandwidth-dominant stage) ---
  k_slice_apply<<<dim3(1024, 4), 256, 0, stream>>>(bgrid, guid, Mm, Mb, thr, slp, gbias, out);
}